// Geo_ODEFunc_18854906429537
// MI455X (gfx1250) — compile-verified
//
#include <hip/hip_runtime.h>
#include <stdint.h>

// out = segment_sum((0.4*vals)[:,None] * x[cols], rows) - x + e
// N_NODES=100000, N_EDGES=625000, EMB=128, fp32.

#define EMB 128
#define EDGES_PER_BLOCK 64   // 8 waves x 8 edges
#define EDGES_PER_WAVE 8
#define THREADS 256

// ---------------------------------------------------------------------------
// Kernel 1: out = e - x   (pure streaming, b128 loads -> v_pk_add_f32 -> b128)
// ---------------------------------------------------------------------------
__global__ void __launch_bounds__(THREADS)
init_out_kernel(const float4* __restrict__ x4,
                const float4* __restrict__ e4,
                float4* __restrict__ out4, int nvec) {
    int i = blockIdx.x * blockDim.x + threadIdx.x;
    if (i < nvec) {
        float4 xv = x4[i];
        float4 ev = e4[i];
        float4 r;
        r.x = ev.x - xv.x;
        r.y = ev.y - xv.y;
        r.z = ev.z - xv.z;
        r.w = ev.w - xv.w;
        out4[i] = r;
    }
}

// ---------------------------------------------------------------------------
// Kernel 2: edge-parallel scatter.
//   - block stages 64 edges' (row, col, val) into LDS via async B128 copies
//   - wave32 layout: lane L owns dims [4L, 4L+3]  (32 lanes * 4 = 128 dims)
//   - full tiles: 8 gathers issued as one clause (8 in-flight 512B L2 hits),
//     then 32 fire-and-forget global_atomic_add_f32
// ---------------------------------------------------------------------------
__device__ __forceinline__ void edge_atomic_add(float* o, float s, float4 xv) {
    const float m0 = s * xv.x;
    const float m1 = s * xv.y;
    const float m2 = s * xv.z;
    const float m3 = s * xv.w;
    asm volatile("global_atomic_add_f32 %0, %1, off"
                 :: "v"(o), "v"(m0) : "memory");
    asm volatile("global_atomic_add_f32 %0, %1, off offset:4"
                 :: "v"(o), "v"(m1) : "memory");
    asm volatile("global_atomic_add_f32 %0, %1, off offset:8"
                 :: "v"(o), "v"(m2) : "memory");
    asm volatile("global_atomic_add_f32 %0, %1, off offset:12"
                 :: "v"(o), "v"(m3) : "memory");
}

__global__ void __launch_bounds__(THREADS)
scatter_kernel(const float* __restrict__ x,
               const float* __restrict__ vals,
               const int*   __restrict__ rows,
               const int*   __restrict__ cols,
               float* __restrict__ out, int n_edges) {
    __shared__ int   sh_rows[EDGES_PER_BLOCK];
    __shared__ int   sh_cols[EDGES_PER_BLOCK];
    __shared__ float sh_vals[EDGES_PER_BLOCK];

    const int tid       = threadIdx.x;
    const int tile_base = blockIdx.x * EDGES_PER_BLOCK;

    // --- async stage of edge metadata: 16 lanes x 16B per array -------------
    if (tid < (EDGES_PER_BLOCK / 4)) {
        int gi   = tile_base + tid * 4;
        int gmax = n_edges - 4;            // clamp source for the tail tile;
        if (gi > gmax) gi = gmax;          // LDS slots stay distinct, unused
        const int*   gr = rows + gi;       // duplicates are guarded below.
        const int*   gc = cols + gi;
        const float* gv = vals + gi;
        uint32_t lr = (uint32_t)(uintptr_t)(&sh_rows[tid * 4]);
        uint32_t lc = (uint32_t)(uintptr_t)(&sh_cols[tid * 4]);
        uint32_t lv = (uint32_t)(uintptr_t)(&sh_vals[tid * 4]);
        asm volatile("global_load_async_to_lds_b128 %0, %1, off"
                     :: "v"(lr), "v"(gr) : "memory");
        asm volatile("global_load_async_to_lds_b128 %0, %1, off"
                     :: "v"(lc), "v"(gc) : "memory");
        asm volatile("global_load_async_to_lds_b128 %0, %1, off"
                     :: "v"(lv), "v"(gv) : "memory");
    }
    // ASYNCcnt is per-wave; waves that issued nothing see 0 (no-op wait).
    asm volatile("s_wait_asynccnt 0x0" ::: "memory");
    __syncthreads();

    const int lane = tid & 31;           // wave32
    const int wave = tid >> 5;           // 0..7
    const int d    = lane * 4;           // this lane's 4 dims
    const int base = wave * EDGES_PER_WAVE;

    if (tile_base + EDGES_PER_BLOCK <= n_edges) {
        // ---- fast path: full tile, no per-edge guards -> deep pipelining ---
        int   rrow[EDGES_PER_WAVE];
        int   rcol[EDGES_PER_WAVE];
        float rs[EDGES_PER_WAVE];
#pragma unroll
        for (int i = 0; i < EDGES_PER_WAVE; ++i) {
            rrow[i] = sh_rows[base + i];
            rcol[i] = sh_cols[base + i];
            rs[i]   = 0.4f * sh_vals[base + i];   // alpha/2 = 0.4
        }
        // 8 coalesced 512B/wave gathers in flight (x is L2-resident: 51MB)
        float4 xv[EDGES_PER_WAVE];
#pragma unroll
        for (int i = 0; i < EDGES_PER_WAVE; ++i) {
            xv[i] = *reinterpret_cast<const float4*>(
                x + (size_t)rcol[i] * EMB + d);
        }
#pragma unroll
        for (int i = 0; i < EDGES_PER_WAVE; ++i) {
            edge_atomic_add(out + (size_t)rrow[i] * EMB + d, rs[i], xv[i]);
        }
    } else {
        // ---- tail tile: guarded path --------------------------------------
        for (int i = 0; i < EDGES_PER_WAVE; ++i) {
            const int local = base + i;
            const int edge  = tile_base + local;
            if (edge >= n_edges) break;
            const int   row = sh_rows[local];
            const int   col = sh_cols[local];
            const float s   = 0.4f * sh_vals[local];
            const float4 xv = *reinterpret_cast<const float4*>(
                x + (size_t)col * EMB + d);
            edge_atomic_add(out + (size_t)row * EMB + d, s, xv);
        }
    }
}

// ---------------------------------------------------------------------------
// Host-side launcher. Inputs (setup_inputs order):
//   0: t (f32, 1)   1: x (f32, N*128)   2: e (f32, N*128)
//   3: hg_vals (f32, E)   4: hg_rows (i32, E)   5: hg_cols (i32, E)
// Output: f32, N*128.
// ---------------------------------------------------------------------------
extern "C" void kernel_launch(void* const* d_in, const int* in_sizes, int n_in,
                              void* d_out, int out_size, void* d_ws, size_t ws_size,
                              hipStream_t stream) {
    (void)n_in; (void)d_ws; (void)ws_size;

    const float* x    = (const float*)d_in[1];
    const float* e    = (const float*)d_in[2];
    const float* vals = (const float*)d_in[3];
    const int*   rows = (const int*)d_in[4];
    const int*   cols = (const int*)d_in[5];
    float*       out  = (float*)d_out;

    const int n_edges = in_sizes[3];
    const int nvec    = out_size / 4;   // float4 count

    init_out_kernel<<<(nvec + THREADS - 1) / THREADS, THREADS, 0, stream>>>(
        (const float4*)x, (const float4*)e, (float4*)out, nvec);

    const int nblocks = (n_edges + EDGES_PER_BLOCK - 1) / EDGES_PER_BLOCK;
    scatter_kernel<<<nblocks, THREADS, 0, stream>>>(
        x, vals, rows, cols, out, n_edges);
}